// HyperLatticeBlock_46291157516385
// MI455X (gfx1250) — compile-verified
//
#include <hip/hip_runtime.h>
#include <math.h>

typedef __attribute__((ext_vector_type(2))) float v2f;
typedef __attribute__((ext_vector_type(8))) float v8f;

// Problem constants (match reference)
#define T_TOK   2048      // B*S
#define DDIM    1024
#define LEXP    48
#define KSEL    4
#define NPAIR   (T_TOK * KSEL)   // 8192
#define MAX_TILES 560            // sum ceil(count_e/16) <= 8192/16 + 48

// ---------------------------------------------------------------------------
// CDNA5 async global->LDS helpers (ASYNCcnt-tracked bulk copies)
// ---------------------------------------------------------------------------
__device__ __forceinline__ void async_issue_b128(unsigned lds_off, const void* g) {
  asm volatile("global_load_async_to_lds_b128 %0, %1, off"
               :: "v"(lds_off), "v"(g) : "memory");
}
__device__ __forceinline__ void wait_async_le2() {
  asm volatile("s_wait_asynccnt 0x2" ::: "memory");
}
__device__ __forceinline__ void wait_async_le0() {
  asm volatile("s_wait_asynccnt 0x0" ::: "memory");
}

// ---------------------------------------------------------------------------
// Shared 16x16-output K=1024 WMMA stream:
//   w_n0 : &W[0][n0] of a row-major [1024][1024] fp32 matrix (B operand)
//   xsT  : LDS tile of A, transposed layout xsT[k*16 + m] (16 rows staged)
//   bsw  : this wave's private 512-float LDS double buffer (16B aligned)
// Weights are double-buffered through LDS via async b128 copies: each chunk is
// a 16x16 fp32 tile (1KB) fetched with two b128 ops (16B/lane), overlapped
// with the previous chunk's 4 WMMAs.
// ---------------------------------------------------------------------------
__device__ __forceinline__ v8f stream_gemm_1024(
    const float* __restrict__ w_n0, const float* __restrict__ xsT,
    float* __restrict__ bsw, const int lane) {
  const int col   = lane & 15;   // N for B/D frags, M for A frag
  const int khalf = lane >> 4;   // 0: K={0,1}, 1: K={2,3}
  const char* wlane = (const char*)w_n0 + (lane & 3) * 16 +
                      (size_t)(lane >> 2) * (DDIM * 4);
  const unsigned lds0 = (unsigned)(size_t)bsw + (unsigned)lane * 16u;
  const float* arow = xsT + 2 * khalf * 16 + col;
  const float* b0 = bsw + 2 * khalf * 16 + col;
  const float* b1 = b0 + 256;

  v8f acc = {};

  auto issue = [&](int c, unsigned boff) {
    const char* g0 = wlane + (size_t)c * (16 * DDIM * 4);
    async_issue_b128(lds0 + boff, g0);
    async_issue_b128(lds0 + boff + 512u, g0 + (size_t)(8 * DDIM * 4));
  };
  auto compute = [&](int c, const float* bb) {
    const float* ar = arow + c * 256;
#pragma unroll
    for (int kl = 0; kl < 16; kl += 4) {
      v2f a, b;
      a.x = ar[(kl + 0) * 16];
      a.y = ar[(kl + 1) * 16];
      b.x = bb[kl * 16];
      b.y = bb[kl * 16 + 16];
      acc = __builtin_amdgcn_wmma_f32_16x16x4_f32(
          false, a, false, b, (short)0, acc, false, false);
    }
  };

  issue(0, 0u);
#pragma unroll 1
  for (int c = 0; c < 62; c += 2) {
    issue(c + 1, 1024u);
    wait_async_le2();
    compute(c, b0);
    issue(c + 2, 0u);
    wait_async_le2();
    compute(c + 1, b1);
  }
  issue(63, 1024u);
  wait_async_le2();
  compute(62, b0);
  wait_async_le0();
  compute(63, b1);
  return acc;
}

// ---------------------------------------------------------------------------
// Kernel 1: gate logits + top-4 + softmax, per token. One block (64 thr)/token.
// ---------------------------------------------------------------------------
__global__ __launch_bounds__(64) void gate_topk_kernel(
    const float* __restrict__ x, const float* __restrict__ gate_w,
    int* __restrict__ idx_out, float* __restrict__ sc_out, int* __restrict__ counts) {
  __shared__ float lx[DDIM];
  __shared__ float lg[LEXP];
  const int t = blockIdx.x;
  const float* xr = x + (size_t)t * DDIM;
  for (int i = threadIdx.x; i < DDIM / 4; i += 64)
    ((float4*)lx)[i] = ((const float4*)xr)[i];
  __syncthreads();
  const int l = threadIdx.x;
  if (l < LEXP) {
    const float* w = gate_w + (size_t)l * DDIM;
    float s = 0.f;
    for (int d = 0; d < DDIM; ++d) s = fmaf(lx[d], w[d], s);
    lg[l] = s;
  }
  __syncthreads();
  if (threadIdx.x == 0) {
    int sel[KSEL]; float sv[KSEL];
    bool used[LEXP];
    for (int j = 0; j < LEXP; ++j) used[j] = false;
    for (int k = 0; k < KSEL; ++k) {
      float best = -INFINITY; int bi = 0;
      for (int j = 0; j < LEXP; ++j)
        if (!used[j] && lg[j] > best) { best = lg[j]; bi = j; }
      used[bi] = true; sel[k] = bi; sv[k] = best;
    }
    const float mx = sv[0];  // descending -> sv[0] is max
    float ev[KSEL]; float den = 0.f;
    for (int k = 0; k < KSEL; ++k) { ev[k] = __expf(sv[k] - mx); den += ev[k]; }
    const float rden = 1.f / den;
    for (int k = 0; k < KSEL; ++k) {
      idx_out[t * KSEL + k] = sel[k];
      sc_out[t * KSEL + k]  = ev[k] * rden;
      atomicAdd(&counts[sel[k]], 1);
    }
  }
}

// ---------------------------------------------------------------------------
// Kernel 2: scan counts -> offsets, build M-tile metadata, zero cursors.
// ---------------------------------------------------------------------------
__global__ __launch_bounds__(64) void plan_kernel(
    const int* __restrict__ counts, int* __restrict__ offsets, int* __restrict__ cursors,
    int* __restrict__ tile_e, int* __restrict__ tile_p, int* __restrict__ tile_n,
    int* __restrict__ num_tiles) {
  if (threadIdx.x < LEXP) cursors[threadIdx.x] = 0;
  if (threadIdx.x == 0) {
    int off = 0, nt = 0;
    for (int e = 0; e < LEXP; ++e) {
      offsets[e] = off;
      const int c = counts[e];
      for (int r = 0; r < c; r += 16) {
        tile_e[nt] = e;
        tile_p[nt] = off + r;
        tile_n[nt] = (c - r < 16) ? (c - r) : 16;
        ++nt;
      }
      off += c;
    }
    offsets[LEXP] = off;
    *num_tiles = nt;
  }
}

// ---------------------------------------------------------------------------
// Kernel 3: scatter pair ids into expert-grouped order.
// ---------------------------------------------------------------------------
__global__ __launch_bounds__(256) void scatter_kernel(
    const int* __restrict__ idx, const int* __restrict__ offsets,
    int* __restrict__ cursors, int* __restrict__ pair_orig) {
  const int p = blockIdx.x * 256 + threadIdx.x;  // 0..NPAIR-1, p = token*4+k
  const int e = idx[p];
  const int pos = offsets[e] + atomicAdd(&cursors[e], 1);
  pair_orig[pos] = p;
}

// ---------------------------------------------------------------------------
// Kernel 4: grouped lattice GEMM, WMMA f32 16x16x4 + async weight streaming.
// Block = 128 thr (4 waves); each wave owns a 16-wide N subtile.
// ---------------------------------------------------------------------------
__global__ __launch_bounds__(128) void lattice_gemm_kernel(
    const float* __restrict__ x, const float* __restrict__ lat_w,
    const int* __restrict__ pair_orig,
    const int* __restrict__ tile_e, const int* __restrict__ tile_p,
    const int* __restrict__ tile_n, const int* __restrict__ num_tiles,
    float* __restrict__ out_pair) {
  __shared__ float xs[DDIM * 16];                    // xs[k*16 + row], 64KB
  __shared__ __align__(16) float bs[4 * 512];        // per-wave dbl buffer, 8KB
  const int tile = blockIdx.y;
  if (tile >= *num_tiles) return;
  const int e      = tile_e[tile];
  const int pstart = tile_p[tile];
  const int nrows  = tile_n[tile];
  const int tid = threadIdx.x;

  // Stage 16 gathered token rows (zero-fill unused rows), transposed.
  {
    const int r = tid >> 3, seg = tid & 7;   // row, 128-float segment
    int token = -1;
    if (r < nrows) token = pair_orig[pstart + r] >> 2;
    for (int i = 0; i < 32; ++i) {
      float4 vv = make_float4(0.f, 0.f, 0.f, 0.f);
      if (token >= 0)
        vv = ((const float4*)(x + (size_t)token * DDIM + seg * 128))[i];
      const int k = seg * 128 + i * 4;
      xs[(k + 0) * 16 + r] = vv.x;
      xs[(k + 1) * 16 + r] = vv.y;
      xs[(k + 2) * 16 + r] = vv.z;
      xs[(k + 3) * 16 + r] = vv.w;
    }
  }
  __syncthreads();

  const int wave = tid >> 5;
  const int lane = tid & 31;
  const int n0   = blockIdx.x * 64 + wave * 16;
  const int col  = lane & 15;

  v8f acc = stream_gemm_1024(lat_w + (size_t)e * DDIM * DDIM + n0,
                             xs, bs + wave * 512, lane);

  // D layout: VGPR r -> M=r (lanes 0-15), M=r+8 (lanes 16-31); N = lane&15.
  // Branchless tail: invalid rows redirect to trash row NPAIR of out_pair.
  const int mhi = (lane >> 4) * 8;
#pragma unroll
  for (int rr = 0; rr < 8; ++rr) {
    const int m = mhi + rr;
    const int gi = pstart + m;
    int po = pair_orig[gi < NPAIR ? gi : NPAIR - 1];
    po = (m < nrows) ? po : NPAIR;
    out_pair[(size_t)po * DDIM + n0 + col] = acc[rr];
  }
}

// ---------------------------------------------------------------------------
// Kernel 5: combine K=4 scored expert outputs per token (fixed order).
// ---------------------------------------------------------------------------
__global__ __launch_bounds__(256) void combine_kernel(
    const float* __restrict__ out_pair, const float* __restrict__ scores,
    float* __restrict__ y_lat) {
  const int t = blockIdx.x;
  const float s0 = scores[t * 4 + 0], s1 = scores[t * 4 + 1];
  const float s2 = scores[t * 4 + 2], s3 = scores[t * 4 + 3];
  const float* p0 = out_pair + (size_t)(t * 4 + 0) * DDIM;
  const float* p1 = out_pair + (size_t)(t * 4 + 1) * DDIM;
  const float* p2 = out_pair + (size_t)(t * 4 + 2) * DDIM;
  const float* p3 = out_pair + (size_t)(t * 4 + 3) * DDIM;
  for (int d = threadIdx.x; d < DDIM; d += 256)
    y_lat[(size_t)t * DDIM + d] = s0 * p0[d] + s1 * p1[d] + s2 * p2[d] + s3 * p3[d];
}

// ---------------------------------------------------------------------------
// Kernel 6: transpose out_w so the out_proj B-fragment loads are coalesced.
// ---------------------------------------------------------------------------
__global__ __launch_bounds__(256) void transpose_kernel(
    const float* __restrict__ out_w, float* __restrict__ out_wt) {
  __shared__ float tile[16][17];
  const int bx = blockIdx.x * 16, by = blockIdx.y * 16;
  const int tx = threadIdx.x & 15, ty = threadIdx.x >> 4;
  tile[ty][tx] = out_w[(size_t)(by + ty) * DDIM + bx + tx];
  __syncthreads();
  out_wt[(size_t)(bx + ty) * DDIM + by + tx] = tile[tx][ty];
}

// ---------------------------------------------------------------------------
// Kernel 7: out_proj WMMA GEMM (async weight streaming) + residual + bias.
// ---------------------------------------------------------------------------
__global__ __launch_bounds__(128) void outproj_kernel(
    const float* __restrict__ y_lat, const float* __restrict__ out_wt,
    const float* __restrict__ x, const float* __restrict__ out_b,
    float* __restrict__ y_full) {
  __shared__ float as[DDIM * 16];                    // as[k*16 + row], 64KB
  __shared__ __align__(16) float bs[4 * 512];        // per-wave dbl buffer, 8KB
  const int t0 = blockIdx.y * 16;
  const int tid = threadIdx.x;
  {
    const int r = tid >> 3, seg = tid & 7;
    const float4* src = (const float4*)(y_lat + (size_t)(t0 + r) * DDIM + seg * 128);
    for (int i = 0; i < 32; ++i) {
      const float4 vv = src[i];
      const int k = seg * 128 + i * 4;
      as[(k + 0) * 16 + r] = vv.x;
      as[(k + 1) * 16 + r] = vv.y;
      as[(k + 2) * 16 + r] = vv.z;
      as[(k + 3) * 16 + r] = vv.w;
    }
  }
  __syncthreads();

  const int wave = tid >> 5;
  const int lane = tid & 31;
  const int n0   = blockIdx.x * 64 + wave * 16;
  const int col  = lane & 15;

  v8f acc = stream_gemm_1024(out_wt + n0, as, bs + wave * 512, lane);

  const float bias = out_b[n0 + col];
  const int mhi = (lane >> 4) * 8;
#pragma unroll
  for (int rr = 0; rr < 8; ++rr) {
    const size_t o = (size_t)(t0 + mhi + rr) * DDIM + n0 + col;
    y_full[o] = x[o] + acc[rr] + bias;
  }
}

// ---------------------------------------------------------------------------
// Kernel 8: per-token LayerNorm.
// ---------------------------------------------------------------------------
__global__ __launch_bounds__(256) void ln_kernel(
    const float* __restrict__ y_full, const float* __restrict__ g,
    const float* __restrict__ b, float* __restrict__ out) {
  __shared__ float red[256];
  const int t = blockIdx.x;
  const float* yr = y_full + (size_t)t * DDIM;
  float v[4]; float s = 0.f;
#pragma unroll
  for (int i = 0; i < 4; ++i) { v[i] = yr[threadIdx.x + i * 256]; s += v[i]; }
  red[threadIdx.x] = s; __syncthreads();
  for (int st = 128; st > 0; st >>= 1) {
    if (threadIdx.x < st) red[threadIdx.x] += red[threadIdx.x + st];
    __syncthreads();
  }
  const float mu = red[0] * (1.f / DDIM);
  __syncthreads();
  float s2 = 0.f;
#pragma unroll
  for (int i = 0; i < 4; ++i) { const float d = v[i] - mu; s2 += d * d; }
  red[threadIdx.x] = s2; __syncthreads();
  for (int st = 128; st > 0; st >>= 1) {
    if (threadIdx.x < st) red[threadIdx.x] += red[threadIdx.x + st];
    __syncthreads();
  }
  const float inv = rsqrtf(red[0] * (1.f / DDIM) + 1e-5f);
#pragma unroll
  for (int i = 0; i < 4; ++i) {
    const int d = threadIdx.x + i * 256;
    out[(size_t)t * DDIM + d] = g[d] * (v[i] - mu) * inv + b[d];
  }
}

// ---------------------------------------------------------------------------
// Host launcher
// ---------------------------------------------------------------------------
extern "C" void kernel_launch(void* const* d_in, const int* in_sizes, int n_in,
                              void* d_out, int out_size, void* d_ws, size_t ws_size,
                              hipStream_t stream) {
  const float* x       = (const float*)d_in[0];
  const float* gate_w  = (const float*)d_in[1];
  const float* lat_w   = (const float*)d_in[2];
  const float* out_w   = (const float*)d_in[3];
  const float* out_b   = (const float*)d_in[4];
  const float* ln_g    = (const float*)d_in[5];
  const float* ln_b    = (const float*)d_in[6];
  float* out = (float*)d_out;

  char* ws = (char*)d_ws;
  size_t o = 0;
  int*   idx       = (int*)(ws + o);   o += (size_t)NPAIR * 4;
  float* scores    = (float*)(ws + o); o += (size_t)NPAIR * 4;
  int*   counts    = (int*)(ws + o);   o += 256;
  int*   offsets   = (int*)(ws + o);   o += 256;
  int*   cursors   = (int*)(ws + o);   o += 256;
  int*   tile_e    = (int*)(ws + o);   o += 4096;
  int*   tile_p    = (int*)(ws + o);   o += 4096;
  int*   tile_n    = (int*)(ws + o);   o += 4096;
  int*   num_tiles = (int*)(ws + o);   o += 256;
  int*   pair_orig = (int*)(ws + o);   o += (size_t)NPAIR * 4;
  float* out_pair  = (float*)(ws + o); o += (size_t)(NPAIR + 1) * DDIM * 4;  // +1 trash row
  float* y_lat     = (float*)(ws + o); o += (size_t)T_TOK * DDIM * 4;   // 8 MB
  float* y_full    = (float*)(ws + o); o += (size_t)T_TOK * DDIM * 4;   // 8 MB
  float* out_wt    = (float*)(ws + o); o += (size_t)DDIM * DDIM * 4;    // 4 MB
  (void)ws_size; (void)in_sizes; (void)n_in; (void)out_size;

  hipMemsetAsync(counts, 0, 256, stream);

  gate_topk_kernel<<<T_TOK, 64, 0, stream>>>(x, gate_w, idx, scores, counts);
  plan_kernel<<<1, 64, 0, stream>>>(counts, offsets, cursors, tile_e, tile_p, tile_n, num_tiles);
  scatter_kernel<<<NPAIR / 256, 256, 0, stream>>>(idx, offsets, cursors, pair_orig);
  lattice_gemm_kernel<<<dim3(DDIM / 64, MAX_TILES), 128, 0, stream>>>(
      x, lat_w, pair_orig, tile_e, tile_p, tile_n, num_tiles, out_pair);
  combine_kernel<<<T_TOK, 256, 0, stream>>>(out_pair, scores, y_lat);
  transpose_kernel<<<dim3(DDIM / 16, DDIM / 16), 256, 0, stream>>>(out_w, out_wt);
  outproj_kernel<<<dim3(DDIM / 64, T_TOK / 16), 128, 0, stream>>>(
      y_lat, out_wt, x, out_b, y_full);
  ln_kernel<<<T_TOK, 256, 0, stream>>>(y_full, ln_g, ln_b, out);
}